// SlidingKVQCache_86268713108247
// MI455X (gfx1250) — compile-verified
//
#include <hip/hip_runtime.h>

// SlidingKVQCache shift for MI455X (gfx1250).
//
// Pure streaming permutation: 805 MB of HBM traffic, 0 FLOPs ->
// HBM-bound, ~34.6 us floor at 23.3 TB/s. We use the CDNA5 async
// global<->LDS copy path (ASYNCcnt-tracked, B128 granularity) so data
// flows memory -> LDS -> memory without a VGPR round trip, and rely on
// massive block-level parallelism (12,288 blocks x 32 KB tiles, >=10
// resident blocks per WGP at 32 KB LDS each) to keep tens of MB in
// flight, well above the ~23 MB bandwidth-delay product of the chip.

#define BLOCK 256
#define STAGE 8                    // float4 per thread per tile
#define CHUNK (BLOCK * STAGE)      // 2048 float4 = 32 KB per block

// Problem shape (fixed by the reference)
#define BB 4
#define HH 32
#define SS 2048
#define DD 128

#define N4     ((unsigned)(BB * HH * SS * (DD / 4)))   // float4 per cache = 8,388,608
#define SD4    ((unsigned)(SS * (DD / 4)))             // float4 per (b,h) slab = 65536
#define LAST4  ((unsigned)((SS - 1) * (DD / 4)))       // shifted-region size = 65504
#define D4     ((unsigned)(DD / 4))                    // 32
#define TOTAL4 (3u * N4)                               // 25,165,824
#define NBLOCKS (TOTAL4 / CHUNK)                       // 12,288 (exact)

__device__ __forceinline__ void async_load_b128(const float4* gsrc, unsigned lds_off) {
    // copies 16 bytes from the global address in gsrc to the LDS address in lds_off, tracked by ASYNCcnt
    asm volatile("global_load_async_to_lds_b128 %0, %1, off"
                 :
                 : "v"(lds_off), "v"(gsrc)
                 : "memory");
}

__device__ __forceinline__ void async_store_b128(float4* gdst, unsigned lds_off) {
    // copies 16 bytes from the LDS address in lds_off to the global address in gdst, tracked by ASYNCcnt
    asm volatile("global_store_async_from_lds_b128 %0, %1, off"
                 :
                 : "v"(gdst), "v"(lds_off)
                 : "memory");
}

__device__ __forceinline__ void wait_async_zero() {
    asm volatile("s_wait_asynccnt 0" ::: "memory");
}

__global__ __launch_bounds__(BLOCK) void SlidingKVQCache_kernel(
    const float* __restrict__ kc, const float* __restrict__ vc,
    const float* __restrict__ qc, const float* __restrict__ tk,
    const float* __restrict__ tv, const float* __restrict__ tq,
    float* __restrict__ out)
{
    __shared__ float4 stage[CHUNK];   // single shared array -> LDS offset 0

    const unsigned t    = threadIdx.x;
    const unsigned base = (unsigned)blockIdx.x * CHUNK;   // out float4 index

    // Which of the three caches this tile belongs to (uniform: N4 % CHUNK == 0)
    const unsigned c = base / N4;
    const float4* cache = (const float4*)(c == 0 ? kc : (c == 1 ? vc : qc));
    const float4* tok   = (const float4*)(c == 0 ? tk : (c == 1 ? tv : tq));
    const unsigned rbase = base - c * N4;   // float4 index within this cache

    // Phase 1: 8 async B128 loads per thread, global -> LDS
    #pragma unroll
    for (int j = 0; j < STAGE; ++j) {
        const unsigned idx  = rbase + (unsigned)j * BLOCK + t;
        const unsigned slab = idx >> 16;          // idx / SD4 (SD4 = 65536)
        const unsigned off  = idx & (SD4 - 1u);   // idx % SD4
        const float4* src = (off < LAST4)
                              ? (cache + (slab * SD4 + off + D4))        // shifted cache row
                              : (tok   + (slab * D4 + (off - LAST4)));   // new token row
        // Low 32 bits of a generic LDS pointer == LDS byte address (ISA 10.2)
        const unsigned loff = (unsigned)(size_t)&stage[(unsigned)j * BLOCK + t];
        async_load_b128(src, loff);
    }

    // Drain loads: LDS tile is now populated for this wave's slots
    wait_async_zero();

    // Phase 2: 8 async B128 stores per thread, LDS -> global
    float4* ob = (float4*)out + base;
    #pragma unroll
    for (int j = 0; j < STAGE; ++j) {
        const unsigned loff = (unsigned)(size_t)&stage[(unsigned)j * BLOCK + t];
        async_store_b128(ob + (unsigned)j * BLOCK + t, loff);
    }

    // Drain stores before wave exit (S_ENDPGM also waits idle, but be explicit)
    wait_async_zero();
}

extern "C" void kernel_launch(void* const* d_in, const int* in_sizes, int n_in,
                              void* d_out, int out_size, void* d_ws, size_t ws_size,
                              hipStream_t stream) {
    const float* kc = (const float*)d_in[0];  // k_cache  [B,H,S,D]
    const float* vc = (const float*)d_in[1];  // v_cache
    const float* qc = (const float*)d_in[2];  // q_cache
    const float* tk = (const float*)d_in[3];  // key      [B,H,1,D]
    const float* tv = (const float*)d_in[4];  // value
    const float* tq = (const float*)d_in[5];  // query
    float* out = (float*)d_out;               // [3, B,H,S,D] concatenated

    SlidingKVQCache_kernel<<<dim3(NBLOCKS), dim3(BLOCK), 0, stream>>>(
        kc, vc, qc, tk, tv, tq, out);
}